// Transformer_layer_53807350284427
// MI455X (gfx1250) — compile-verified
//
#include <hip/hip_runtime.h>
#include <math.h>

// ---------------------------------------------------------------------------
// 2-layer TransformerConv GNN for gfx1250 (MI455X), f32 end-to-end.
// Dense projections use V_WMMA_F32_16X16X4_F32 (bias folded into C);
// edge phase is L2-resident gathers + float atomics (102MB feature table
// fits the 192MB L2, so random per-edge access never hits HBM in steady state).
// ---------------------------------------------------------------------------

typedef float v2f __attribute__((ext_vector_type(2)));
typedef float v8f __attribute__((ext_vector_type(8)));

#define WPB 8  // waves per block (8 * wave32 = 256 threads)

__device__ __forceinline__ void atomicMaxF(float* addr, float val) {
    // monotone-bits trick; init = 0xFFFFFFFF (max as uint, -1 as int,
    // so any real value wins on first update)
    if (val >= 0.0f) atomicMax((int*)addr, __float_as_int(val));
    else             atomicMin((unsigned int*)addr, __float_as_uint(val));
}

// ---------------------------------------------------------------------------
// Pack weights: W1pad [4][256] = [Wq1|Wk1|Wv1|Ws1] (K=3 padded to 4, zero row)
//               b1cat [256];  W2pad [64][32] = [Wq2|Wk2|Wv2|Ws2|pad]; b2cat[32]
// ---------------------------------------------------------------------------
__global__ void pack_weights(const float* Wq1, const float* bq1, const float* Wk1, const float* bk1,
                             const float* Wv1, const float* bv1, const float* Ws1, const float* bs1,
                             const float* Wq2, const float* bq2, const float* Wk2, const float* bk2,
                             const float* Wv2, const float* bv2, const float* Ws2, const float* bs2,
                             float* W1pad, float* b1cat, float* W2pad, float* b2cat) {
    int t = threadIdx.x + blockIdx.x * blockDim.x;
    if (t < 4 * 256) {
        int k = t / 256, c = t % 256, mat = c / 64, cc = c % 64;
        const float* W = (mat == 0) ? Wq1 : (mat == 1) ? Wk1 : (mat == 2) ? Wv1 : Ws1;
        W1pad[t] = (k < 3) ? W[k * 64 + cc] : 0.0f;
    }
    if (t < 256) {
        int mat = t / 64;
        const float* B = (mat == 0) ? bq1 : (mat == 1) ? bk1 : (mat == 2) ? bv1 : bs1;
        b1cat[t] = B[t % 64];
    }
    if (t < 64 * 32) {
        int k = t / 32, c = t % 32;
        float v = 0.0f;
        if (c < 24) {
            int mat = c / 6;
            const float* W = (mat == 0) ? Wq2 : (mat == 1) ? Wk2 : (mat == 2) ? Wv2 : Ws2;
            v = W[k * 6 + (c % 6)];
        }
        W2pad[t] = v;
    }
    if (t < 32) {
        float v = 0.0f;
        if (t < 24) {
            int mat = t / 6;
            const float* B = (mat == 0) ? bq2 : (mat == 1) ? bk2 : (mat == 2) ? bv2 : bs2;
            v = B[t % 6];
        }
        b2cat[t] = v;
    }
}

// ---------------------------------------------------------------------------
// Layer-1 projections: proj1[N][256] = x[N][3(pad4)] @ W1pad[4][256] + b1cat
// One wave = one 16-node tile; 16 column tiles; one K=4 f32 WMMA each.
// Bias folded into C init; full tiles take an unguarded coalesced store path.
// ---------------------------------------------------------------------------
__global__ __launch_bounds__(256) void proj1_kernel(const float* __restrict__ x,
                                                    const float* __restrict__ W1pad,
                                                    const float* __restrict__ b1cat,
                                                    float* __restrict__ proj1,
                                                    int N, int ntiles) {
    int wave = blockIdx.x * WPB + (threadIdx.x >> 5);
    int lane = threadIdx.x & 31;
    if (wave >= ntiles) return;           // wave-uniform; EXEC stays all-1 below
    int row = lane & 15;                  // A row / D column within tile
    int kh  = lane >> 4;                  // lane-half selects K-half
    int base = wave * 16;
    int node = base + row;
    int nc = node < N ? node : N - 1;
    // A 16x4 (M x K): lanes0-15 hold K=0,1; lanes16-31 hold K=2,3 (K=3 is pad)
    v2f a;
    a.x = x[(size_t)nc * 3 + 2 * kh];
    a.y = (kh == 0) ? x[(size_t)nc * 3 + 1] : 0.0f;

    bool full = (base + 16 <= N);         // wave-uniform
    if (full) {
        for (int ct = 0; ct < 16; ++ct) {
            int col = ct * 16 + row;
            v2f b;
            b.x = W1pad[(2 * kh + 0) * 256 + col];
            b.y = W1pad[(2 * kh + 1) * 256 + col];
            float bias = b1cat[col];
            v8f c = {bias, bias, bias, bias, bias, bias, bias, bias};
            c = __builtin_amdgcn_wmma_f32_16x16x4_f32(false, a, false, b, (short)0, c, false, false);
            float* outp = proj1 + (size_t)(base + 8 * kh) * 256 + col;
#pragma unroll
            for (int j = 0; j < 8; ++j) outp[(size_t)j * 256] = c[j];
        }
    } else {
        for (int ct = 0; ct < 16; ++ct) {
            int col = ct * 16 + row;
            v2f b;
            b.x = W1pad[(2 * kh + 0) * 256 + col];
            b.y = W1pad[(2 * kh + 1) * 256 + col];
            float bias = b1cat[col];
            v8f c = {bias, bias, bias, bias, bias, bias, bias, bias};
            c = __builtin_amdgcn_wmma_f32_16x16x4_f32(false, a, false, b, (short)0, c, false, false);
#pragma unroll
            for (int j = 0; j < 8; ++j) {
                int outn = base + j + 8 * kh;
                if (outn < N) proj1[(size_t)outn * 256 + col] = c[j];
            }
        }
    }
}

// ---------------------------------------------------------------------------
// Layer-1 edge attention. proj1 layout: [q 0:64 | k 64:128 | v 128:192 | s 192:256]
// thread = (edge, head)
// ---------------------------------------------------------------------------
__global__ void l1_logits(const int* __restrict__ ei, const float* __restrict__ proj1,
                          float* __restrict__ logits1, float* __restrict__ m1, int E) {
    long long tid = (long long)blockIdx.x * blockDim.x + threadIdx.x;
    if (tid >= (long long)E * 4) return;
    int e = (int)(tid >> 2), h = (int)(tid & 3);
    int src = ei[e], dst = ei[E + e];
    const float4* q = (const float4*)(proj1 + (size_t)dst * 256 + h * 16);
    const float4* k = (const float4*)(proj1 + (size_t)src * 256 + 64 + h * 16);
    float d = 0.0f;
#pragma unroll
    for (int i = 0; i < 4; ++i) {
        float4 qq = q[i], kk = k[i];
        d += qq.x * kk.x + qq.y * kk.y + qq.z * kk.z + qq.w * kk.w;
    }
    d *= 0.25f; // 1/sqrt(16)
    logits1[(size_t)e * 4 + h] = d;
    atomicMaxF(m1 + (size_t)dst * 4 + h, d);
}

__global__ void l1_accum(const int* __restrict__ ei, const float* __restrict__ proj1,
                         const float* __restrict__ logits1, const float* __restrict__ m1,
                         float* __restrict__ den1, float* __restrict__ agg1, int E) {
    long long tid = (long long)blockIdx.x * blockDim.x + threadIdx.x;
    if (tid >= (long long)E * 4) return;
    int e = (int)(tid >> 2), h = (int)(tid & 3);
    int src = ei[e], dst = ei[E + e];
    float a = __expf(logits1[(size_t)e * 4 + h] - m1[(size_t)dst * 4 + h]);
    atomicAdd(den1 + (size_t)dst * 4 + h, a);
    const float4* v = (const float4*)(proj1 + (size_t)src * 256 + 128 + h * 16);
    float* out = agg1 + (size_t)dst * 64 + h * 16;
#pragma unroll
    for (int i = 0; i < 4; ++i) {
        float4 vv = v[i];
        atomicAdd(out + i * 4 + 0, a * vv.x);
        atomicAdd(out + i * 4 + 1, a * vv.y);
        atomicAdd(out + i * 4 + 2, a * vv.z);
        atomicAdd(out + i * 4 + 3, a * vv.w);
    }
}

// ---------------------------------------------------------------------------
// Fused: h1 = relu(agg1/denom + skip) -> LDS tile -> K=64 WMMA chain into
// proj2[N][32] = [q2 0:6 | k2 6:12 | v2 12:18 | s2 18:24 | pad]
// ---------------------------------------------------------------------------
__global__ __launch_bounds__(256) void proj2_kernel(const float* __restrict__ agg1,
                                                    const float* __restrict__ den1,
                                                    const float* __restrict__ proj1,
                                                    const float* __restrict__ W2pad,
                                                    const float* __restrict__ b2cat,
                                                    float* __restrict__ proj2,
                                                    int N, int ntiles) {
    __shared__ float h1[WPB][16][65]; // 65: avoid 16-way bank conflict on col reads
    int ws   = threadIdx.x >> 5;
    int lane = threadIdx.x & 31;
    int wave = blockIdx.x * WPB + ws;
    bool active = wave < ntiles;      // wave-uniform
    int base = wave * 16;
    bool full = (base + 16 <= N);     // wave-uniform
    if (active) {
        if (full) {
            for (int i = 0; i < 32; ++i) {
                int flat = lane * 32 + i;
                int r = flat >> 6, c = flat & 63;
                int n = base + r;
                float den = den1[(size_t)n * 4 + (c >> 4)];
                float v = agg1[(size_t)n * 64 + c] / (den + 1e-16f) + proj1[(size_t)n * 256 + 192 + c];
                h1[ws][r][c] = fmaxf(v, 0.0f);
            }
        } else {
            for (int i = 0; i < 32; ++i) {
                int flat = lane * 32 + i;
                int r = flat >> 6, c = flat & 63;
                int n = base + r;
                float v = 0.0f;
                if (n < N) {
                    float den = den1[(size_t)n * 4 + (c >> 4)];
                    v = agg1[(size_t)n * 64 + c] / (den + 1e-16f) + proj1[(size_t)n * 256 + 192 + c];
                    v = fmaxf(v, 0.0f);
                }
                h1[ws][r][c] = v;
            }
        }
    }
    __syncthreads();
    if (!active) return;

    int row = lane & 15, kh = lane >> 4;
    float bias0 = b2cat[row], bias1 = b2cat[16 + row];
    v8f c0 = {bias0, bias0, bias0, bias0, bias0, bias0, bias0, bias0};
    v8f c1 = {bias1, bias1, bias1, bias1, bias1, bias1, bias1, bias1};
    for (int kk = 0; kk < 16; ++kk) {
        int k0 = kk * 4 + 2 * kh;
        v2f a;
        a.x = h1[ws][row][k0 + 0];
        a.y = h1[ws][row][k0 + 1];
        v2f b0, b1v;
        b0.x  = W2pad[(k0 + 0) * 32 + row];
        b0.y  = W2pad[(k0 + 1) * 32 + row];
        b1v.x = W2pad[(k0 + 0) * 32 + 16 + row];
        b1v.y = W2pad[(k0 + 1) * 32 + 16 + row];
        c0 = __builtin_amdgcn_wmma_f32_16x16x4_f32(false, a, false, b0,  (short)0, c0, false, false);
        c1 = __builtin_amdgcn_wmma_f32_16x16x4_f32(false, a, false, b1v, (short)0, c1, false, false);
    }
    if (full) {
        float* outp = proj2 + (size_t)(base + 8 * kh) * 32 + row;
#pragma unroll
        for (int j = 0; j < 8; ++j) {
            outp[(size_t)j * 32]      = c0[j];
            outp[(size_t)j * 32 + 16] = c1[j];
        }
    } else {
#pragma unroll
        for (int j = 0; j < 8; ++j) {
            int n = base + j + 8 * kh;
            if (n < N) {
                proj2[(size_t)n * 32 + row]      = c0[j];
                proj2[(size_t)n * 32 + 16 + row] = c1[j];
            }
        }
    }
}

// ---------------------------------------------------------------------------
// Layer-2 edge attention (heads=1, d=6)
// ---------------------------------------------------------------------------
__global__ void l2_logits(const int* __restrict__ ei, const float* __restrict__ proj2,
                          float* __restrict__ logits2, float* __restrict__ m2, int E) {
    int e = blockIdx.x * blockDim.x + threadIdx.x;
    if (e >= E) return;
    int src = ei[e], dst = ei[E + e];
    const float* q = proj2 + (size_t)dst * 32;
    const float* k = proj2 + (size_t)src * 32 + 6;
    float d = 0.0f;
#pragma unroll
    for (int j = 0; j < 6; ++j) d += q[j] * k[j];
    d *= 0.4082482904638631f; // 1/sqrt(6)
    logits2[e] = d;
    atomicMaxF(m2 + dst, d);
}

__global__ void l2_accum(const int* __restrict__ ei, const float* __restrict__ proj2,
                         const float* __restrict__ logits2, const float* __restrict__ m2,
                         float* __restrict__ den2, float* __restrict__ agg2, int E) {
    int e = blockIdx.x * blockDim.x + threadIdx.x;
    if (e >= E) return;
    int src = ei[e], dst = ei[E + e];
    float a = __expf(logits2[e] - m2[dst]);
    atomicAdd(den2 + dst, a);
    const float* v = proj2 + (size_t)src * 32 + 12;
    float* out = agg2 + (size_t)dst * 8;
#pragma unroll
    for (int j = 0; j < 6; ++j) atomicAdd(out + j, a * v[j]);
}

// ---------------------------------------------------------------------------
// h2 = agg2/denom + skip2 ; mean-pool accumulate per graph
// ---------------------------------------------------------------------------
__global__ void node_pool(const float* __restrict__ agg2, const float* __restrict__ den2,
                          const float* __restrict__ proj2, const int* __restrict__ batch,
                          float* __restrict__ pooled, float* __restrict__ cnt, int N) {
    int n = blockIdx.x * blockDim.x + threadIdx.x;
    if (n >= N) return;
    float inv = 1.0f / (den2[n] + 1e-16f);
    int g = batch[n];
#pragma unroll
    for (int j = 0; j < 6; ++j)
        atomicAdd(pooled + (size_t)g * 6 + j,
                  agg2[(size_t)n * 8 + j] * inv + proj2[(size_t)n * 32 + 18 + j]);
    atomicAdd(cnt + g, 1.0f);
}

__global__ void pool_logsoftmax(const float* __restrict__ pooled, const float* __restrict__ cnt,
                                float* __restrict__ out, int G) {
    int g = blockIdx.x * blockDim.x + threadIdx.x;
    if (g >= G) return;
    float c = fmaxf(cnt[g], 1.0f);
    float xs[6], m = -INFINITY;
#pragma unroll
    for (int j = 0; j < 6; ++j) { xs[j] = pooled[(size_t)g * 6 + j] / c; m = fmaxf(m, xs[j]); }
    float s = 0.0f;
#pragma unroll
    for (int j = 0; j < 6; ++j) s += __expf(xs[j] - m);
    float lse = __logf(s);
#pragma unroll
    for (int j = 0; j < 6; ++j) out[(size_t)g * 6 + j] = xs[j] - m - lse;
}

// ---------------------------------------------------------------------------
extern "C" void kernel_launch(void* const* d_in, const int* in_sizes, int n_in,
                              void* d_out, int out_size, void* d_ws, size_t ws_size,
                              hipStream_t stream) {
    const float* x     = (const float*)d_in[0];
    const int*   ei    = (const int*)d_in[1];
    const int*   batch = (const int*)d_in[2];
    int N = in_sizes[0] / 3;
    int E = in_sizes[1] / 2;
    int G = out_size / 6;

    float* ws = (float*)d_ws;
    size_t o = 0;
    float* proj1   = ws + o; o += (size_t)N * 256;
    float* m1      = ws + o; o += (size_t)N * 4;
    float* den1    = ws + o; o += (size_t)N * 4;
    float* agg1    = ws + o; o += (size_t)N * 64;
    float* logits1 = ws + o; o += (size_t)E * 4;
    float* proj2   = ws + o; o += (size_t)N * 32;
    float* m2      = ws + o; o += (size_t)N;
    float* den2    = ws + o; o += (size_t)N;
    float* agg2    = ws + o; o += (size_t)N * 8;
    float* logits2 = ws + o; o += (size_t)E;
    float* pooled  = ws + o; o += (size_t)G * 6;
    float* cnt     = ws + o; o += (size_t)G;          // contiguous with pooled
    float* W1pad   = ws + o; o += 4 * 256;
    float* b1cat   = ws + o; o += 256;
    float* W2pad   = ws + o; o += 64 * 32;
    float* b2cat   = ws + o; o += 32;

    // deterministic per-call init (capture-safe async memsets)
    (void)hipMemsetAsync(den1,   0,    (size_t)N * 4  * sizeof(float), stream);
    (void)hipMemsetAsync(agg1,   0,    (size_t)N * 64 * sizeof(float), stream);
    (void)hipMemsetAsync(den2,   0,    (size_t)N      * sizeof(float), stream);
    (void)hipMemsetAsync(agg2,   0,    (size_t)N * 8  * sizeof(float), stream);
    (void)hipMemsetAsync(pooled, 0,    ((size_t)G * 6 + G) * sizeof(float), stream);
    (void)hipMemsetAsync(m1,     0xFF, (size_t)N * 4  * sizeof(float), stream); // "-inf" bits
    (void)hipMemsetAsync(m2,     0xFF, (size_t)N      * sizeof(float), stream);

    pack_weights<<<8, 256, 0, stream>>>(
        (const float*)d_in[3],  (const float*)d_in[4],  (const float*)d_in[5],  (const float*)d_in[6],
        (const float*)d_in[7],  (const float*)d_in[8],  (const float*)d_in[9],  (const float*)d_in[10],
        (const float*)d_in[11], (const float*)d_in[12], (const float*)d_in[13], (const float*)d_in[14],
        (const float*)d_in[15], (const float*)d_in[16], (const float*)d_in[17], (const float*)d_in[18],
        W1pad, b1cat, W2pad, b2cat);

    int ntiles = (N + 15) / 16;
    int pblocks = (ntiles + WPB - 1) / WPB;
    proj1_kernel<<<pblocks, 256, 0, stream>>>(x, W1pad, b1cat, proj1, N, ntiles);

    long long t1 = (long long)E * 4;
    int eb4 = (int)((t1 + 255) / 256);
    l1_logits<<<eb4, 256, 0, stream>>>(ei, proj1, logits1, m1, E);
    l1_accum <<<eb4, 256, 0, stream>>>(ei, proj1, logits1, m1, den1, agg1, E);

    proj2_kernel<<<pblocks, 256, 0, stream>>>(agg1, den1, proj1, W2pad, b2cat, proj2, N, ntiles);

    int eb = (E + 255) / 256;
    l2_logits<<<eb, 256, 0, stream>>>(ei, proj2, logits2, m2, E);
    l2_accum <<<eb, 256, 0, stream>>>(ei, proj2, logits2, m2, den2, agg2, E);

    node_pool<<<(N + 255) / 256, 256, 0, stream>>>(agg2, den2, proj2, batch, pooled, cnt, N);
    pool_logsoftmax<<<1, 64, 0, stream>>>(pooled, cnt, (float*)d_out, G);
}